// SoftDTW_47991964566278
// MI455X (gfx1250) — compile-verified
//
#include <hip/hip_runtime.h>

typedef __attribute__((ext_vector_type(2))) float v2f;
typedef __attribute__((ext_vector_type(4))) float v4f;
typedef __attribute__((ext_vector_type(8))) float v8f;

#define TLEN 1024
#define DDIM 64
#define BATCH 32
#define BIGF 1.0e10f

// ---------------------------------------------------------------------------
// Kernel 0: per-row squared norms  x2[b,t] = sum_d x[b,t,d]^2  (and same for y)
// ---------------------------------------------------------------------------
__global__ void __launch_bounds__(256) sqnorm_kernel(
    const float* __restrict__ x, const float* __restrict__ y,
    float* __restrict__ x2, float* __restrict__ y2, int nrows)
{
    int r = blockIdx.x * blockDim.x + threadIdx.x;
    if (r >= nrows) return;
    const v4f* xr = reinterpret_cast<const v4f*>(x + (size_t)r * DDIM);
    const v4f* yr = reinterpret_cast<const v4f*>(y + (size_t)r * DDIM);
    float sx = 0.0f, sy = 0.0f;
#pragma unroll
    for (int i = 0; i < DDIM / 4; ++i) {
        v4f a = xr[i];
        v4f b = yr[i];
        sx += a[0]*a[0] + a[1]*a[1] + a[2]*a[2] + a[3]*a[3];
        sy += b[0]*b[0] + b[1]*b[1] + b[2]*b[2] + b[3]*b[3];
    }
    x2[r] = sx;
    y2[r] = sy;
}

// ---------------------------------------------------------------------------
// Kernel 1: WMMA cost matrix.  One wave computes one 16x16 tile of
//   Dxy[b,i,j] = max(0, x2[i] + y2[j] - 2 * sum_d x[i,d]*y[j,d])
// using V_WMMA_F32_16X16X4_F32 (exact f32), K-loop over D=64 in steps of 4.
//
// f32 A-matrix 16x4 layout: lane L, VGPR v -> A[L%16][2*(L/16)+v]
// f32 B-matrix 4x16 layout: lane L, VGPR v -> B[2*(L/16)+v][L%16]
// Since B = Y^T and Y is row-major, both operands load as contiguous float2
// from row (tile_base + L%16), columns (k0 + 2*(L/16)).
// C/D layout: lane L, VGPR r -> element (r + 8*(L/16), L%16).
// ---------------------------------------------------------------------------
__global__ void __launch_bounds__(128) dxy_wmma_kernel(
    const float* __restrict__ x, const float* __restrict__ y,
    const float* __restrict__ x2, const float* __restrict__ y2,
    float* __restrict__ Dxy, int b0, int nb)
{
    const int wavesPerBlock = 4;
    int wave = blockIdx.x * wavesPerBlock + (threadIdx.x >> 5);
    int lane = threadIdx.x & 31;

    const int tilesPerBatch = (TLEN / 16) * (TLEN / 16); // 4096
    int b = wave / tilesPerBatch;
    if (b >= nb) return;
    int rem = wave % tilesPerBatch;
    int ti = rem >> 6;        // 64 tile-cols per row of tiles
    int tj = rem & 63;
    int row0 = ti * 16;
    int col0 = tj * 16;
    int gb = b0 + b;

    int l16  = lane & 15;
    int half = lane >> 4;

    const float* xp = x + ((size_t)gb * TLEN + row0 + l16) * DDIM + half * 2;
    const float* yp = y + ((size_t)gb * TLEN + col0 + l16) * DDIM + half * 2;

    v8f acc = {0.f, 0.f, 0.f, 0.f, 0.f, 0.f, 0.f, 0.f};
#pragma unroll
    for (int k0 = 0; k0 < DDIM; k0 += 4) {
        v2f av = *reinterpret_cast<const v2f*>(xp + k0);
        v2f bv = *reinterpret_cast<const v2f*>(yp + k0);
        acc = __builtin_amdgcn_wmma_f32_16x16x4_f32(
            /*neg_a=*/false, av, /*neg_b=*/false, bv,
            /*c_mod=*/(short)0, acc, /*reuse_a=*/false, /*reuse_b=*/false);
    }

    float yn = y2[(size_t)gb * TLEN + col0 + l16];
    const float* xnp = x2 + (size_t)gb * TLEN + row0 + half * 8;
    float* outp = Dxy + ((size_t)b * TLEN + row0 + half * 8) * TLEN + col0 + l16;
#pragma unroll
    for (int r = 0; r < 8; ++r) {
        float c = xnp[r] + yn - 2.0f * acc[r];
        c = fmaxf(c, 0.0f);
        outp[(size_t)r * TLEN] = c;
    }
}

// ---------------------------------------------------------------------------
// Kernel 2: soft-DTW anti-diagonal DP.  One 1024-thread (32-wave) workgroup
// per batch; three rolling diagonals of length T+1 in LDS; 2T-1 barriered
// steps.  gamma = 1.
//   R[i][j] = cost[i-1][j-1] + softmin(R[i-1][j-1], R[i-1][j], R[i][j-1])
// Diagonal k holds d[idx] = R[idx][k-idx].
//
// Thread tid owns idx = tid+1 (idx 0 is statically BIG; thread 0 refreshes
// it in the rotating buffer).  The cost for (idx, j=k-idx) lives at
// C[tid*T + (k-tid-2)], i.e. a unit-stride per-thread stream: it is
// prefetched one diagonal ahead, with the load issued BEFORE the barrier so
// its latency overlaps the barrier + next step's LDS/softmin chain.
// ---------------------------------------------------------------------------
__global__ void __launch_bounds__(1024) softdtw_dp_kernel(
    const float* __restrict__ Dxy, float* __restrict__ out, int b0, int nb)
{
    int b = blockIdx.x;
    if (b >= nb) return;

    __shared__ float Dbuf[3][TLEN + 1];
    int tid = threadIdx.x;
    int idx = tid + 1;                      // 1..1024

    // init: diag k=0 -> [0]=0, rest BIG; diag k=1 -> all BIG
    for (int i = tid; i <= TLEN; i += 1024) {
        Dbuf[0][i] = (i == 0) ? 0.0f : BIGF;
        Dbuf[1][i] = BIGF;
    }
    __syncthreads();

    const float* cp = Dxy + (size_t)b * TLEN * TLEN + (size_t)tid * TLEN; // row tid
    int pm2 = 0, pm1 = 1, pc = 2;

    // prefetch cost for k=2: col = 2 - idx - 1 = -tid  (valid only for tid 0)
    float cost_next = (tid == 0) ? cp[0] : 0.0f;

    for (int k = 2; k <= 2 * TLEN; ++k) {
        float cost = cost_next;

        int j = k - idx;                         // column in R
        bool valid = (j >= 1) && (j <= TLEN);

        float a  = Dbuf[pm2][idx - 1];           // R[i-1][j-1]
        float bb = Dbuf[pm1][idx - 1];           // R[i-1][j]
        float c  = Dbuf[pm1][idx];               // R[i][j-1]

        float m  = fminf(a, fminf(bb, c));
        float sm = m - __logf(__expf(m - a) + __expf(m - bb) + __expf(m - c));

        Dbuf[pc][idx] = valid ? (cost + sm) : BIGF;
        if (tid == 0) Dbuf[pc][0] = BIGF;        // idx 0 is always invalid

        // prefetch cost for k+1 (independent of LDS; issued before barrier)
        int coln = k - 1 - tid;                  // (k+1) - idx - 1
        cost_next = 0.0f;
        if (coln >= 0 && coln < TLEN) cost_next = cp[coln];

        __syncthreads();
        int t = pm2; pm2 = pm1; pm1 = pc; pc = t;
    }

    if (tid == 0) out[b0 + b] = Dbuf[pm1][TLEN]; // diag k=2T, idx=T
}

// ---------------------------------------------------------------------------
// Host launcher.  Chunks batches so that the per-chunk cost matrix fits in the
// provided workspace (pure host arithmetic + stream launches: graph-safe).
// ---------------------------------------------------------------------------
extern "C" void kernel_launch(void* const* d_in, const int* in_sizes, int n_in,
                              void* d_out, int out_size, void* d_ws, size_t ws_size,
                              hipStream_t stream) {
    const float* x = (const float*)d_in[0];
    const float* y = (const float*)d_in[1];
    float* out = (float*)d_out;

    const size_t rows = (size_t)BATCH * TLEN;            // 32768
    float* x2 = (float*)d_ws;
    float* y2 = x2 + rows;
    size_t normBytes = 2 * rows * sizeof(float);         // 256 KB
    size_t dxyOff = (normBytes + 255) & ~(size_t)255;
    float* Dxy = (float*)((char*)d_ws + dxyOff);

    size_t perBatch = (size_t)TLEN * TLEN * sizeof(float); // 4 MB
    size_t avail = (ws_size > dxyOff) ? (ws_size - dxyOff) : 0;
    int nbMax = (int)(avail / perBatch);
    if (nbMax > BATCH) nbMax = BATCH;
    if (nbMax < 1) nbMax = 1;

    sqnorm_kernel<<<(int)((rows + 255) / 256), 256, 0, stream>>>(x, y, x2, y2, (int)rows);

    const int tilesPerBatch = (TLEN / 16) * (TLEN / 16); // 4096
    for (int bch = 0; bch < BATCH; bch += nbMax) {
        int nb = BATCH - bch;
        if (nb > nbMax) nb = nbMax;
        int waves = nb * tilesPerBatch;
        int blocks = (waves + 3) / 4; // 4 waves (128 threads) per block
        dxy_wmma_kernel<<<blocks, 128, 0, stream>>>(x, y, x2, y2, Dxy, bch, nb);
        softdtw_dp_kernel<<<nb, 1024, 0, stream>>>(Dxy, out, bch, nb);
    }
}